// CenterDirectionLoss_77627238908096
// MI455X (gfx1250) — compile-verified
//
#include <hip/hip_runtime.h>

// ---------------------------------------------------------------------------
// CenterDirectionLoss for MI455X (gfx1250, wave32).
// Memory-bound: ~190 MB HBM traffic -> ~8 us at 23.3 TB/s. f32 throughout.
// WMMA (V_WMMA_F32_16X16X4_F32, all-ones B) used as exact deterministic
// 32-lane f32 wave reduction. Non-temporal loads for single-use f32 planes.
// ---------------------------------------------------------------------------

typedef float v2f __attribute__((ext_vector_type(2)));
typedef float v4f __attribute__((ext_vector_type(4)));
typedef float v8f __attribute__((ext_vector_type(8)));
typedef int   v4i __attribute__((ext_vector_type(4)));

#define BB        16
#define HH        768
#define WW        768
#define PP        (HH * WW)        // 589824 pixels per batch-plane
#define NIDS      128
#define NCHUNK    64               // main-pass blocks per batch
#define CHUNK_PIX (PP / NCHUNK)    // 9216 pixels per block
#define MAIN_IT   (CHUNK_PIX / 4 / 256)  // 9 float4 iters per thread
#define NHBLK     32               // histogram blocks per batch
#define HB_PIX    (PP / NHBLK)     // 18432
#define HIST_IT   (HB_PIX / 4 / 256)     // 18 int4 iters per thread

// Workspace layout (32-bit words):
//   [0,      2048)  counts   (u32)  B*NIDS
//   [2048,   4096)  invnorm  (f32)  B*NIDS
//   [4096,   6144)  partials (f32)  B*NCHUNK*2  (ls, lc)
#define WS_COUNTS   0
#define WS_INVNORM  2048
#define WS_PARTIALS 4096

// Exact f32 sum across the 32 lanes of a wave using two WMMAs.
// A-matrix layout (documented): lanes 0-15 -> rows 0-15 {K0=VGPR0,K1=VGPR1},
// lanes 16-31 -> rows 0-15 {K2=VGPR0,K3=VGPR1}. B = all-ones (layout-proof).
// WMMA1: D.Vj (any lane group) = x[j+8*hi] + x[j+8*hi+16]   (rowsum_m)
// WMMA2 on (V0..3 summed, V4..7 summed): every D element = total sum.
__device__ __forceinline__ float wave_sum_wmma(float x) {
  v2f ones; ones.x = 1.0f; ones.y = 1.0f;
  v8f c = {};
  v2f a; a.x = x; a.y = 0.0f;
  v8f d = __builtin_amdgcn_wmma_f32_16x16x4_f32(
      false, a, false, ones, (short)0, c, false, false);
  v2f a2;
  a2.x = (d[0] + d[1]) + (d[2] + d[3]);
  a2.y = (d[4] + d[5]) + (d[6] + d[7]);
  v8f d2 = __builtin_amdgcn_wmma_f32_16x16x4_f32(
      false, a2, false, ones, (short)0, c, false, false);
  return d2[0];
}

__global__ __launch_bounds__(256) void zero_counts_kernel(unsigned* counts) {
  int t = blockIdx.x * 256 + threadIdx.x;
  if (t < BB * NIDS) counts[t] = 0u;
}

__global__ __launch_bounds__(256) void hist_kernel(const int* __restrict__ inst,
                                                   unsigned* __restrict__ counts) {
  __shared__ unsigned s_cnt[NIDS];
  const int b = blockIdx.x / NHBLK;
  const int c = blockIdx.x % NHBLK;
  const int t = threadIdx.x;
  if (t < NIDS) s_cnt[t] = 0u;
  __syncthreads();
  const v4i* ii = (const v4i*)(inst + (size_t)b * PP) + (size_t)c * (HB_PIX / 4);
  for (int i = 0; i < HIST_IT; ++i) {
    v4i id = ii[i * 256 + t];
    atomicAdd(&s_cnt[id.x], 1u);
    atomicAdd(&s_cnt[id.y], 1u);
    atomicAdd(&s_cnt[id.z], 1u);
    atomicAdd(&s_cnt[id.w], 1u);
  }
  __syncthreads();
  if (t < NIDS) atomicAdd(&counts[b * NIDS + t], s_cnt[t]);
}

// Single block: global scalars + per-(b,id) inverse norms.
__global__ __launch_bounds__(256) void weights_kernel(const unsigned* __restrict__ counts,
                                                      float* __restrict__ invnorm) {
  __shared__ unsigned sh_ni[256];
  __shared__ unsigned sh_bg[256];
  __shared__ float f_ni2, f_bg2;
  const int t = threadIdx.x;
  unsigned ni = 0u, bg = 0u;
  for (int i = t; i < BB * NIDS; i += 256) {
    unsigned cnt = counts[i];
    if ((i & (NIDS - 1)) == 0) bg += cnt;
    else                       ni += (cnt > 0u) ? 1u : 0u;
  }
  sh_ni[t] = ni; sh_bg[t] = bg;
  __syncthreads();
  for (int s = 128; s > 0; s >>= 1) {
    if (t < s) { sh_ni[t] += sh_ni[t + s]; sh_bg[t] += sh_bg[t + s]; }
    __syncthreads();
  }
  if (t == 0) {
    f_ni2 = 2.0f * (float)sh_ni[0];   // 2 * num_instances
    f_bg2 = 2.0f * (float)sh_bg[0];   // 2 * num_bg_pixels
  }
  __syncthreads();
  for (int i = t; i < BB * NIDS; i += 256) {
    float cnt  = (float)counts[i];
    float norm = ((i & (NIDS - 1)) == 0) ? f_bg2 : fmaxf(cnt, 1.0f) * f_ni2;
    invnorm[i] = 1.0f / norm;
  }
}

__global__ __launch_bounds__(256) void main_loss_kernel(const float* __restrict__ pred,
                                                        const int* __restrict__ inst,
                                                        const float* __restrict__ gt,
                                                        const float* __restrict__ invnorm,
                                                        float* __restrict__ partials) {
  __shared__ float s_inv[NIDS];
  __shared__ float s_ls[8];
  __shared__ float s_lc[8];
  const int b = blockIdx.x / NCHUNK;
  const int c = blockIdx.x % NCHUNK;
  const int t = threadIdx.x;
  if (t < NIDS) s_inv[t] = invnorm[b * NIDS + t];
  __syncthreads();

  const size_t q4 = (size_t)c * (CHUNK_PIX / 4);
  const v4i* ii = (const v4i*)(inst + (size_t)b * PP) + q4;
  const v4f* ps = (const v4f*)(pred + (size_t)b * 2 * PP) + q4;              // pred sin
  const v4f* pc = (const v4f*)(pred + (size_t)b * 2 * PP + PP) + q4;         // pred cos
  const v4f* gs = (const v4f*)(gt + (size_t)b * 5 * PP + 2 * (size_t)PP) + q4;  // gt sin
  const v4f* gc = (const v4f*)(gt + (size_t)b * 5 * PP + 3 * (size_t)PP) + q4;  // gt cos

  float als = 0.0f, alc = 0.0f;
#pragma unroll
  for (int i = 0; i < MAIN_IT; ++i) {
    const int off = i * 256 + t;
    v4i id = ii[off];                                     // cached (reused by hist pass / L2)
    v4f a  = __builtin_nontemporal_load(&ps[off]);        // single-use streams: NT
    v4f p  = __builtin_nontemporal_load(&pc[off]);
    v4f g0 = __builtin_nontemporal_load(&gs[off]);
    v4f g1 = __builtin_nontemporal_load(&gc[off]);
    float w0 = s_inv[id.x], w1 = s_inv[id.y], w2 = s_inv[id.z], w3 = s_inv[id.w];
    als = fmaf(w0, fabsf(a.x - g0.x), als);
    als = fmaf(w1, fabsf(a.y - g0.y), als);
    als = fmaf(w2, fabsf(a.z - g0.z), als);
    als = fmaf(w3, fabsf(a.w - g0.w), als);
    alc = fmaf(w0, fabsf(p.x - g1.x), alc);
    alc = fmaf(w1, fabsf(p.y - g1.y), alc);
    alc = fmaf(w2, fabsf(p.z - g1.z), alc);
    alc = fmaf(w3, fabsf(p.w - g1.w), alc);
  }

  // Exact wave32 reduction via WMMA (EXEC is all-ones here: uniform control flow).
  float wls = wave_sum_wmma(als);
  float wlc = wave_sum_wmma(alc);
  const int wave = t >> 5, lane = t & 31;
  if (lane == 0) { s_ls[wave] = wls; s_lc[wave] = wlc; }
  __syncthreads();
  if (t == 0) {
    float ls = 0.0f, lc = 0.0f;
#pragma unroll
    for (int wv = 0; wv < 8; ++wv) { ls += s_ls[wv]; lc += s_lc[wv]; }
    partials[(size_t)(b * NCHUNK + c) * 2 + 0] = ls;
    partials[(size_t)(b * NCHUNK + c) * 2 + 1] = lc;
  }
}

// Outputs: [loss, loss_dir_total, loss_centers, loss_sin, loss_cos], each 16 f32.
__global__ __launch_bounds__(32) void finalize_kernel(const float* __restrict__ partials,
                                                      float* __restrict__ out) {
  const int b = threadIdx.x;
  if (b < BB) {
    float ls = 0.0f, lc = 0.0f;
    for (int c = 0; c < NCHUNK; ++c) {     // fixed order -> deterministic
      ls += partials[(size_t)(b * NCHUNK + c) * 2 + 0];
      lc += partials[(size_t)(b * NCHUNK + c) * 2 + 1];
    }
    const float tot = ls + lc;
    out[ 0 + b] = tot;    // loss
    out[16 + b] = tot;    // loss_direction_total
    out[32 + b] = 0.0f;   // loss_centers
    out[48 + b] = ls;     // loss_sin
    out[64 + b] = lc;     // loss_cos
  }
}

extern "C" void kernel_launch(void* const* d_in, const int* in_sizes, int n_in,
                              void* d_out, int out_size, void* d_ws, size_t ws_size,
                              hipStream_t stream) {
  const float* pred = (const float*)d_in[0];   // (16, 2, 768, 768) f32
  const int*   inst = (const int*)d_in[1];     // (16, 768, 768) i32
  // d_in[2] = labels: unused (W_FG == W_BG == 1.0)
  const float* gt   = (const float*)d_in[3];   // (16, 5, 768, 768) f32

  unsigned* ws_counts  = (unsigned*)d_ws + WS_COUNTS;
  float*    ws_invnorm = (float*)d_ws + WS_INVNORM;
  float*    ws_part    = (float*)d_ws + WS_PARTIALS;
  float*    out        = (float*)d_out;

  zero_counts_kernel<<<(BB * NIDS + 255) / 256, 256, 0, stream>>>(ws_counts);
  hist_kernel<<<BB * NHBLK, 256, 0, stream>>>(inst, ws_counts);
  weights_kernel<<<1, 256, 0, stream>>>(ws_counts, ws_invnorm);
  main_loss_kernel<<<BB * NCHUNK, 256, 0, stream>>>(pred, inst, gt, ws_invnorm, ws_part);
  finalize_kernel<<<1, 32, 0, stream>>>(ws_part, out);
}